// SeparateHiddenGCAEEncoder_16286515987229
// MI455X (gfx1250) — compile-verified
//
#include <hip/hip_runtime.h>

typedef __attribute__((ext_vector_type(2))) float v2f;
typedef __attribute__((ext_vector_type(8))) float v8f;

static inline int ceilDiv(int a, int b) { return (a + b - 1) / b; }

// ---------------------------------------------------------------------------
// Degree / normalization helpers
// ---------------------------------------------------------------------------
__global__ void k_fill1(float* __restrict__ p, int n) {
  int t = blockIdx.x * blockDim.x + threadIdx.x;
  if (t < n) p[t] = 1.0f;               // self-loop contributes 1 to every degree
}

__global__ void k_deg(float* __restrict__ deg, const int* __restrict__ dst, int e) {
  int t = blockIdx.x * blockDim.x + threadIdx.x;
  if (t < e) atomicAdd(&deg[dst[t]], 1.0f);
}

__global__ void k_rsqrt_inplace(float* __restrict__ p, int n) {
  int t = blockIdx.x * blockDim.x + threadIdx.x;
  if (t < n) p[t] = rsqrtf(p[t]);       // deg > 0 always (self-loops)
}

__global__ void k_tanh_inplace(float* __restrict__ p, int n) {
  int t = blockIdx.x * blockDim.x + threadIdx.x;
  if (t < n) p[t] = tanhf(p[t]);
}

// ---------------------------------------------------------------------------
// FP32 WMMA GEMM: out[M,OUTC] = A[M,K] @ W[K,OUTC], exact fp32 via
// V_WMMA_F32_16X16X4_F32. One wave computes one 16x16 output tile; a block
// of OUTC*2 threads (OUTC/16 waves) covers 16 rows x OUTC cols. A-tile is
// staged in LDS (shared by all waves in the block), W stays in L2 (tiny).
// EXEC is all-1s at every WMMA (no divergence before the stores).
// ---------------------------------------------------------------------------
template <int K, int OUTC>
__global__ void gemm_wmma_f32(const float* __restrict__ A,
                              const float* __restrict__ W,
                              float* __restrict__ out, int nRows) {
  constexpr int LDA = K + 1;            // pad to dodge LDS bank conflicts
  __shared__ float As[16 * LDA];

  const int tid  = threadIdx.x;
  const int row0 = blockIdx.x << 4;

  for (int idx = tid; idx < 16 * K; idx += OUTC * 2) {
    int r = idx / K, c = idx - r * K;
    int rr = row0 + r; if (rr >= nRows) rr = nRows - 1;   // clamp (N%16==0 here)
    As[r * LDA + c] = A[(size_t)rr * K + c];
  }
  __syncthreads();

  const int lane = tid & 31;
  const int wave = tid >> 5;
  const int half = lane >> 4;           // 0: lanes 0-15 (K k..k+1), 1: lanes 16-31 (K k+2..k+3)
  const int l16  = lane & 15;
  const int col0 = wave << 4;

  const float* Wp = W + col0 + l16;     // this lane's output column of W
  v8f acc = {};

#pragma unroll
  for (int k0 = 0; k0 < K; k0 += 4) {
    const int kk = k0 + (half << 1);
    v2f a, b;
    a.x = As[l16 * LDA + kk];           // A: M=l16, K=kk   (VGPR0)
    a.y = As[l16 * LDA + kk + 1];       //            K=kk+1 (VGPR1)
    b.x = Wp[(size_t)kk * OUTC];        // B: N=l16, K=kk
    b.y = Wp[(size_t)(kk + 1) * OUTC];
    acc = __builtin_amdgcn_wmma_f32_16x16x4_f32(
        /*neg_a=*/false, a, /*neg_b=*/false, b,
        /*c_mod=*/(short)0, acc, /*reuse_a=*/false, /*reuse_b=*/false);
  }

#pragma unroll
  for (int r = 0; r < 8; ++r) {         // D: VGPR r -> rows r (lanes 0-15) / r+8 (lanes 16-31)
    int row = row0 + r + (half << 3);
    if (row < nRows) out[(size_t)row * OUTC + col0 + l16] = acc[r];
  }
}

// ---------------------------------------------------------------------------
// acc[i, off+c] = bias[c] + xl[i,c] * dinv[i]^2   (bias + self-loop term)
// ---------------------------------------------------------------------------
__global__ void gcn_init(const float* __restrict__ xl, const float* __restrict__ dinv,
                         const float* __restrict__ bias, float* __restrict__ acc,
                         int dimShift, int accStride, int accOff, int nNodes) {
  int t = blockIdx.x * blockDim.x + threadIdx.x;
  int i = t >> dimShift;
  int c = t & ((1 << dimShift) - 1);
  if (i >= nNodes) return;
  float di = dinv[i];
  acc[(size_t)i * accStride + accOff + c] =
      bias[c] + xl[((size_t)i << dimShift) + c] * di * di;
}

// ---------------------------------------------------------------------------
// acc[dst, off+c] += xl[src,c] * dinv[src]*dinv[dst]  (one thread per
// (edge, channel); channel-major -> coalesced gathers and atomics)
// ---------------------------------------------------------------------------
__global__ void gcn_scatter(const float* __restrict__ xl, const float* __restrict__ dinv,
                            const int* __restrict__ srcI, const int* __restrict__ dstI,
                            float* __restrict__ acc,
                            int dimShift, int accStride, int accOff, int nEdges) {
  int t = blockIdx.x * blockDim.x + threadIdx.x;
  int e = t >> dimShift;
  int c = t & ((1 << dimShift) - 1);
  if (e >= nEdges) return;
  int s = srcI[e], d = dstI[e];
  float w = dinv[s] * dinv[d];
  atomicAdd(&acc[(size_t)d * accStride + accOff + c],
            xl[((size_t)s << dimShift) + c] * w);
}

// ---------------------------------------------------------------------------
// Launch: 4 GCN convs.  Workspace layout (floats):
//   dinv : N
//   xl   : N*128   (GEMM output of the current conv)
//   h    : N*256   (concat hidden; reused as conv3 accumulator + conv4 input)
// Total 385*N floats ~= 154 MB.
// ---------------------------------------------------------------------------
extern "C" void kernel_launch(void* const* d_in, const int* in_sizes, int n_in,
                              void* d_out, int out_size, void* d_ws, size_t ws_size,
                              hipStream_t stream) {
  const float* feature   = (const float*)d_in[0];
  const float* condition = (const float*)d_in[1];
  const int*   edge      = (const int*)d_in[2];   // [2, E] flattened
  const float* W_f2h = (const float*)d_in[3];
  const float* b_f2h = (const float*)d_in[4];
  const float* W_c2h = (const float*)d_in[5];
  const float* b_c2h = (const float*)d_in[6];
  const float* W_h2h = (const float*)d_in[7];
  const float* b_h2h = (const float*)d_in[8];
  const float* W_h2l = (const float*)d_in[9];
  const float* b_h2l = (const float*)d_in[10];

  const int N = in_sizes[0] / 256;   // FEATURE_DIM
  const int E = in_sizes[2] / 2;
  const int* src = edge;             // edge_index[0]
  const int* dst = edge + E;         // edge_index[1]

  float* dinv = (float*)d_ws;
  float* xl   = dinv + N;
  float* h    = xl + (size_t)N * 128;
  float* out  = (float*)d_out;

  const int mTiles = ceilDiv(N, 16);

  // ---- degrees -> dinv ----
  k_fill1<<<ceilDiv(N, 256), 256, 0, stream>>>(dinv, N);
  k_deg<<<ceilDiv(E, 256), 256, 0, stream>>>(dinv, dst, E);
  k_rsqrt_inplace<<<ceilDiv(N, 256), 256, 0, stream>>>(dinv, N);

  // ---- conv1: tanh(gcn(feature @ W_f2h)) -> h[:, 0:128] ----
  gemm_wmma_f32<256, 128><<<mTiles, 256, 0, stream>>>(feature, W_f2h, xl, N);
  gcn_init<<<ceilDiv(N << 7, 256), 256, 0, stream>>>(xl, dinv, b_f2h, h, 7, 256, 0, N);
  gcn_scatter<<<ceilDiv(E << 7, 256), 256, 0, stream>>>(xl, dinv, src, dst, h, 7, 256, 0, E);

  // ---- conv2: tanh(gcn(condition @ W_c2h)) -> h[:, 128:256] ----
  gemm_wmma_f32<64, 128><<<mTiles, 256, 0, stream>>>(condition, W_c2h, xl, N);
  gcn_init<<<ceilDiv(N << 7, 256), 256, 0, stream>>>(xl, dinv, b_c2h, h, 7, 256, 128, N);
  gcn_scatter<<<ceilDiv(E << 7, 256), 256, 0, stream>>>(xl, dinv, src, dst, h, 7, 256, 128, E);

  // tanh over the whole concat hidden at once
  k_tanh_inplace<<<ceilDiv(N * 256, 256), 256, 0, stream>>>(h, N * 256);

  // ---- conv3: tanh(gcn(h @ W_h2h)); accumulator reuses h as [N,128] ----
  gemm_wmma_f32<256, 128><<<mTiles, 256, 0, stream>>>(h, W_h2h, xl, N);
  gcn_init<<<ceilDiv(N << 7, 256), 256, 0, stream>>>(xl, dinv, b_h2h, h, 7, 128, 0, N);
  gcn_scatter<<<ceilDiv(E << 7, 256), 256, 0, stream>>>(xl, dinv, src, dst, h, 7, 128, 0, E);
  k_tanh_inplace<<<ceilDiv(N * 128, 256), 256, 0, stream>>>(h, N * 128);

  // ---- conv4: z = gcn(h @ W_h2l) -> d_out (no tanh) ----
  gemm_wmma_f32<128, 64><<<mTiles, 128, 0, stream>>>(h, W_h2l, xl, N);
  gcn_init<<<ceilDiv(N << 6, 256), 256, 0, stream>>>(xl, dinv, b_h2l, out, 6, 64, 0, N);
  gcn_scatter<<<ceilDiv(E << 6, 256), 256, 0, stream>>>(xl, dinv, src, dst, out, 6, 64, 0, E);
}